// SpectralFeatureExtractor_74818330296384
// MI455X (gfx1250) — compile-verified
//
#include <hip/hip_runtime.h>
#include <hip/hip_bf16.h>

// ---------------- problem constants ----------------
#define BB   16
#define NN   1024
#define FF   512
#define EE   16384
#define HH   2048
#define NPAIR (NN/2)          // 512 independent Jacobi pairs per round
#define NSWEEPS 8

typedef float v2f __attribute__((ext_vector_type(2)));
typedef float v8f __attribute__((ext_vector_type(8)));
typedef int   v4i __attribute__((ext_vector_type(4)));

// ---------------- CDNA5 async global->LDS support ----------------
#if __has_builtin(__builtin_amdgcn_global_load_async_to_lds_b128) && \
    __has_builtin(__builtin_amdgcn_global_load_async_to_lds_b32)
#define HAVE_ASYNC_LDS 1
#else
#define HAVE_ASYNC_LDS 0
#endif

#if __has_builtin(__builtin_amdgcn_s_wait_asynccnt)
#define WAIT_ASYNC(n) __builtin_amdgcn_s_wait_asynccnt(n)
#else
#define WAIT_ASYNC(n) asm volatile("s_wait_asynccnt %0" ::"i"(n) : "memory")
#endif

// global (AS1 / __device__) and LDS (AS3) typed pointers per the builtin
// signatures reported by clang:
//   b128: (int4 AS1*, int4 AS3*, Imm offset, Imm cpol)
//   b32 : (int  AS1*, int  AS3*, Imm offset, Imm cpol)
typedef __attribute__((address_space(1))) v4i g_v4i;
typedef __attribute__((address_space(3))) v4i l_v4i;
typedef __attribute__((address_space(1))) int g_i32;
typedef __attribute__((address_space(3))) int l_i32;

__device__ __forceinline__ void async_ld_b128(const void* g, void* l) {
#if HAVE_ASYNC_LDS
    __builtin_amdgcn_global_load_async_to_lds_b128((g_v4i*)g, (l_v4i*)l, 0, 0);
#else
    (void)g; (void)l;
#endif
}
__device__ __forceinline__ void async_ld_b32(const void* g, void* l) {
#if HAVE_ASYNC_LDS
    __builtin_amdgcn_global_load_async_to_lds_b32((g_i32*)g, (l_i32*)l, 0, 0);
#else
    (void)g; (void)l;
#endif
}

// V_WMMA_F32_16X16X4_F32: D(16x16,f32) = A(16x4,f32) * B(4x16,f32) + C
// args: (neg_a, A, neg_b, B, c_mod, C, reuse_a, reuse_b)
__device__ __forceinline__ v8f wmma4(v2f a, v2f b, v8f c) {
    return __builtin_amdgcn_wmma_f32_16x16x4_f32(false, a, false, b, (short)0, c,
                                                 false, false);
}

__device__ __forceinline__ float block_reduce_sum(float v, float* sbuf) {
    int t = threadIdx.x;
    sbuf[t] = v;
    __syncthreads();
    for (int off = blockDim.x >> 1; off > 0; off >>= 1) {
        if (t < off) sbuf[t] += sbuf[t + off];
        __syncthreads();
    }
    float r = sbuf[0];
    __syncthreads();
    return r;
}

// ---------------- small utility kernels ----------------
__global__ __launch_bounds__(256) void zero_kernel(float* p, int n) {
    int i = blockIdx.x * 256 + threadIdx.x;
    if (i < n) p[i] = 0.0f;
}

// A[b][src][dst] += w[b][e]   (duplicates summed via atomics, as in .at[].add)
__global__ __launch_bounds__(256) void scatter_kernel(const int* __restrict__ ei,
                                                      const float* __restrict__ ew,
                                                      float* __restrict__ Abuf) {
    int idx = blockIdx.x * 256 + threadIdx.x;
    if (idx >= BB * EE) return;
    int b = idx >> 14;            // / EE
    int e = idx & (EE - 1);
    int src = ei[e];
    int dst = ei[EE + e];
    atomicAdd(&Abuf[((size_t)b * NN + src) * NN + dst],
              ew[(size_t)b * EE + e]);
}

// dis[b][i] = rsqrt(rowsum(A[b][i][:]) + 1e-8)
__global__ __launch_bounds__(256) void degree_kernel(const float* __restrict__ Abuf,
                                                     float* __restrict__ dis) {
    int row = blockIdx.x;  // b*NN + i
    const float* ar = Abuf + (size_t)row * NN;
    float s = 0.0f;
    for (int j = threadIdx.x; j < NN; j += 256) s += ar[j];
    __shared__ float sb[256];
    s = block_reduce_sum(s, sb);
    if (threadIdx.x == 0) dis[row] = rsqrtf(s + 1e-8f);
}

// L = I - D^-1/2 A D^-1/2, symmetrized:  L[i][j] = d_ij - 0.5*dis_i*dis_j*(A_ij + A_ji)
__global__ __launch_bounds__(256) void build_L_kernel(const float* __restrict__ Abuf,
                                                      const float* __restrict__ dis,
                                                      float* __restrict__ Lbuf) {
    int ij = blockIdx.x * 256 + threadIdx.x;   // 0 .. NN*NN-1
    int b  = blockIdx.y;
    int i = ij >> 10;
    int j = ij & (NN - 1);
    size_t base = (size_t)b * NN * NN;
    float aij = Abuf[base + ((size_t)i << 10) + j];
    float aji = Abuf[base + ((size_t)j << 10) + i];
    float dd  = dis[(b << 10) + i] * dis[(b << 10) + j];
    Lbuf[base + ij] = ((i == j) ? 1.0f : 0.0f) - 0.5f * dd * (aij + aji);
}

// ---------------- batched parallel cyclic Jacobi eigensolver ----------------
// One 1024-thread workgroup per batch. Working matrix Aw (destroyed),
// eigenvector accumulator Vw (init I). Ends with bitonic sort of eigenvalues
// (ascending, to match eigh ordering used by the MLP) and a column permute of
// Vw into Uout.
__global__ __launch_bounds__(1024) void jacobi_evd_kernel(float* __restrict__ Aw_all,
                                                          float* __restrict__ Vw_all,
                                                          float* __restrict__ Uout_all,
                                                          float* __restrict__ eig_all) {
    const int b = blockIdx.x;
    const int t = threadIdx.x;
    float* Aw   = Aw_all   + (size_t)b * NN * NN;
    float* Vw   = Vw_all   + (size_t)b * NN * NN;
    float* Uout = Uout_all + (size_t)b * NN * NN;

    __shared__ float sc[NPAIR];
    __shared__ float ss[NPAIR];
    __shared__ int   sp[NPAIR];
    __shared__ int   sq[NPAIR];
    __shared__ float skey[NN];
    __shared__ int   sidx[NN];

    // V = I
    for (int idx = t; idx < NN * NN; idx += 1024) {
        int i = idx >> 10, j = idx & (NN - 1);
        Vw[idx] = (i == j) ? 1.0f : 0.0f;
    }
    __syncthreads();

    for (int sweep = 0; sweep < NSWEEPS; ++sweep) {
        for (int r = 0; r < NN - 1; ++r) {
            // --- phase 0: tournament pairing + rotation angles ---
            if (t < NPAIR) {
                const int n1 = NN - 1;
                int p = (t == 0) ? 0 : 1 + ((t - 1 + r) % n1);
                int q = 1 + ((NN - 2 - t + r) % n1);
                if (p > q) { int tmp = p; p = q; q = tmp; }
                float app = Aw[(size_t)p * NN + p];
                float aqq = Aw[(size_t)q * NN + q];
                float apq = Aw[(size_t)p * NN + q];
                float c = 1.0f, s = 0.0f;
                if (fabsf(apq) > 1e-20f) {
                    float theta = 0.5f * (aqq - app) / apq;
                    float tt = copysignf(1.0f, theta) /
                               (fabsf(theta) + sqrtf(1.0f + theta * theta));
                    c = rsqrtf(1.0f + tt * tt);
                    s = tt * c;
                }
                sp[t] = p; sq[t] = q; sc[t] = c; ss[t] = s;
            }
            __syncthreads();

            // --- phase 1: column rotations on A and V (A*J, V*J) ---
            for (int task = t; task < NN * NPAIR; task += 1024) {
                int i = task >> 9;          // row
                int k = task & (NPAIR - 1); // pair
                int p = sp[k], q = sq[k];
                float c = sc[k], s = ss[k];
                size_t ip = (size_t)i * NN + p, iq = (size_t)i * NN + q;
                float aip = Aw[ip], aiq = Aw[iq];
                Aw[ip] = c * aip - s * aiq;
                Aw[iq] = s * aip + c * aiq;
                float vip = Vw[ip], viq = Vw[iq];
                Vw[ip] = c * vip - s * viq;
                Vw[iq] = s * vip + c * viq;
            }
            __syncthreads();

            // --- phase 2: row rotations on A (J^T * A) ---
            for (int task = t; task < NPAIR * NN; task += 1024) {
                int k = task >> 10;        // pair
                int j = task & (NN - 1);   // col
                int p = sp[k], q = sq[k];
                float c = sc[k], s = ss[k];
                size_t pj = (size_t)p * NN + j, qj = (size_t)q * NN + j;
                float apj = Aw[pj], aqj = Aw[qj];
                Aw[pj] = c * apj - s * aqj;
                Aw[qj] = s * apj + c * aqj;
            }
            __syncthreads();
        }
    }

    // --- sort eigenpairs ascending (bitonic in LDS) ---
    skey[t] = Aw[(size_t)t * NN + t];
    sidx[t] = t;
    __syncthreads();
    for (int ksz = 2; ksz <= NN; ksz <<= 1) {
        for (int j = ksz >> 1; j > 0; j >>= 1) {
            int ixj = t ^ j;
            if (ixj > t) {
                bool up = ((t & ksz) == 0);
                float k1 = skey[t], k2 = skey[ixj];
                bool sw = up ? (k1 > k2) : (k1 < k2);
                if (sw) {
                    skey[t] = k2; skey[ixj] = k1;
                    int ti = sidx[t]; sidx[t] = sidx[ixj]; sidx[ixj] = ti;
                }
            }
            __syncthreads();
        }
    }
    eig_all[(size_t)b * NN + t] = skey[t];
    __syncthreads();

    // --- permute eigenvector columns into output: U[n][m] = V[n][sidx[m]] ---
    for (int task = t; task < NN * NN; task += 1024) {
        int n = task >> 10;
        int m = task & (NN - 1);
        Uout[(size_t)n * NN + m] = Vw[(size_t)n * NN + sidx[m]];
    }
}

// ---------------- LayerNorm over feature dim ----------------
__global__ __launch_bounds__(256) void layernorm_kernel(const float* __restrict__ x,
                                                        const float* __restrict__ gamma,
                                                        const float* __restrict__ beta,
                                                        float* __restrict__ xn) {
    int row = blockIdx.x;  // b*NN + n
    const float* xr = x + (size_t)row * FF;
    float s = 0.0f, s2 = 0.0f;
    for (int f = threadIdx.x; f < FF; f += 256) {
        float v = xr[f];
        s += v; s2 += v * v;
    }
    __shared__ float sb[256];
    s  = block_reduce_sum(s, sb);
    s2 = block_reduce_sum(s2, sb);
    float mu  = s * (1.0f / FF);
    float var = s2 * (1.0f / FF) - mu * mu;
    float inv = rsqrtf(var + 1e-5f);
    for (int f = threadIdx.x; f < FF; f += 256) {
        xn[(size_t)row * FF + f] = (xr[f] - mu) * inv * gamma[f] + beta[f];
    }
}

// ---------------- eigenvalue MLP: 16xK @ KxNout (+bias, opt ReLU) ----------------
// One wave (32 threads) per 16-column output tile; A is the 16-row matrix.
template <bool RELU>
__global__ __launch_bounds__(32) void mlp_gemm_kernel(const float* __restrict__ A,
                                                      const float* __restrict__ W,
                                                      const float* __restrict__ bias,
                                                      float* __restrict__ Out,
                                                      int K, int Nout) {
    const int col0 = blockIdx.x * 16;
    const int lane = threadIdx.x;
    const int lrow = lane & 15;
    const int half = lane >> 4;
    v8f acc = {};
    for (int k = 0; k < K; k += 4) {
        int ka = k + 2 * half;
        v2f a, bfr;
        a.x  = A[(size_t)lrow * K + ka];
        a.y  = A[(size_t)lrow * K + ka + 1];
        bfr.x = W[(size_t)ka * Nout + col0 + lrow];
        bfr.y = W[(size_t)(ka + 1) * Nout + col0 + lrow];
        acc = wmma4(a, bfr, acc);
    }
    float bv = bias[col0 + lrow];
#pragma unroll
    for (int r = 0; r < 8; ++r) {
        int row = r + 8 * half;
        float v = acc[r] + bv;
        if (RELU) v = fmaxf(v, 0.0f);
        Out[(size_t)row * Nout + col0 + lrow] = v;
    }
}

// ---------------- batched spectral-projection GEMM (fp32 WMMA) ---------------
// C[b,m,f] = sum_k opA(U)[m,k] * g(X[b,k,f])    M=K=NN, Fdim=FF
//   TRANS_A : opA(U)[m,k] = U[k*NN+m]    (U^T)
//   RELU_B  : g = relu                   (fuses xa = relu(xn))
//   SCALE_B : g(X)[k,f] = X[k,f]*ev[b,k] (fuses x1 * ev_feat)
//   ADD_RES : epilogue adds Res[b,m,f]   (residual xn)
// Macro tile 64(M) x 128(F), KC=16, 8 waves, 2x2 WMMA tiles per wave.
// Staging uses GLOBAL_LOAD_ASYNC_TO_LDS with double buffering (ASYNCcnt),
// falling back to synchronous LDS staging if the builtins are unavailable.
// ReLU / ev-scaling are applied at fragment-read time (async copies are raw).
template <bool TRANS_A, bool RELU_B, bool SCALE_B, bool ADD_RES>
__global__ __launch_bounds__(256) void gemm_wmma_kernel(const float* __restrict__ U,
                                                        const float* __restrict__ X,
                                                        const float* __restrict__ ev,
                                                        const float* __restrict__ Res,
                                                        float* __restrict__ C) {
    const int tid  = threadIdx.x;
    const int lane = tid & 31;
    const int wave = tid >> 5;
    const int lrow = lane & 15;
    const int half = lane >> 4;
    const int wrow = wave >> 2;  // 0..1
    const int wcol = wave & 3;   // 0..3
    const int m0 = blockIdx.x * 64;
    const int f0 = blockIdx.y * 128;
    const int b  = blockIdx.z;

    const float* Ug  = U + (size_t)b * NN * NN;
    const float* Xg  = X + (size_t)b * NN * FF;
    const float* evg = SCALE_B ? (ev + (size_t)b * NN) : nullptr;
    const float* Rg  = ADD_RES ? (Res + (size_t)b * NN * FF) : nullptr;
    float*       Cg  = C + (size_t)b * NN * FF;

    // +4 padding keeps every row stride a multiple of 16B (b128 LDS dst align)
    __shared__ __align__(16) float As[2][16][64 + 4];
    __shared__ __align__(16) float Bs[2][16][128 + 4];
    __shared__ float Es[2][16];

    v8f acc[2][2] = {};

#if HAVE_ASYNC_LDS
    // ---- async staging: per wave 3 (TRANS) / 6 (non-TRANS) async ops/chunk
    auto stage = [&](int kc, int nb) {
        if (TRANS_A) {
            // global contiguous along m -> one b128 per thread
            int k = tid >> 4, mb = tid & 15;
            async_ld_b128(&Ug[(size_t)(kc + k) * NN + (m0 + mb * 4)],
                          &As[nb][k][mb * 4]);
        } else {
            // global contiguous along k, LDS along m -> 4 x b32 per thread
#pragma unroll
            for (int j = 0; j < 4; ++j) {
                int e = tid + j * 256;
                int m = e >> 4, k = e & 15;
                async_ld_b32(&Ug[(size_t)(m0 + m) * NN + (kc + k)],
                             &As[nb][k][m]);
            }
        }
#pragma unroll
        for (int j = 0; j < 2; ++j) {
            int e = tid + j * 256;   // 512 b128 ops cover 16x128 floats
            int k = e >> 5, fb = e & 31;
            async_ld_b128(&Xg[(size_t)(kc + k) * FF + (f0 + fb * 4)],
                          &Bs[nb][k][fb * 4]);
        }
        if (SCALE_B && tid < 16) Es[nb][tid] = evg[kc + tid];
    };
#else
    // ---- synchronous staging fallback (raw copies; transforms at read)
    auto stage = [&](int kc, int nb) {
        if (TRANS_A) {
            for (int idx = tid; idx < 16 * 64; idx += 256) {
                int k = idx >> 6, m = idx & 63;
                As[nb][k][m] = Ug[(size_t)(kc + k) * NN + (m0 + m)];
            }
        } else {
            for (int idx = tid; idx < 16 * 64; idx += 256) {
                int m = idx >> 4, k = idx & 15;
                As[nb][k][m] = Ug[(size_t)(m0 + m) * NN + (kc + k)];
            }
        }
        for (int idx = tid; idx < 16 * 128; idx += 256) {
            int k = idx >> 7, f = idx & 127;
            Bs[nb][k][f] = Xg[(size_t)(kc + k) * FF + (f0 + f)];
        }
        if (SCALE_B && tid < 16) Es[nb][tid] = evg[kc + tid];
    };
#endif

    auto compute = [&](int cur) {
#pragma unroll
        for (int kk = 0; kk < 16; kk += 4) {
            int kb = kk + 2 * half;
            v2f a0, a1, b0, b1;
            a0.x = As[cur][kb][wrow * 32 + lrow];
            a0.y = As[cur][kb + 1][wrow * 32 + lrow];
            a1.x = As[cur][kb][wrow * 32 + 16 + lrow];
            a1.y = As[cur][kb + 1][wrow * 32 + 16 + lrow];
            b0.x = Bs[cur][kb][wcol * 32 + lrow];
            b0.y = Bs[cur][kb + 1][wcol * 32 + lrow];
            b1.x = Bs[cur][kb][wcol * 32 + 16 + lrow];
            b1.y = Bs[cur][kb + 1][wcol * 32 + 16 + lrow];
            if (RELU_B) {
                b0.x = fmaxf(b0.x, 0.0f); b0.y = fmaxf(b0.y, 0.0f);
                b1.x = fmaxf(b1.x, 0.0f); b1.y = fmaxf(b1.y, 0.0f);
            }
            if (SCALE_B) {
                float e0 = Es[cur][kb], e1 = Es[cur][kb + 1];
                b0.x *= e0; b0.y *= e1;
                b1.x *= e0; b1.y *= e1;
            }
            acc[0][0] = wmma4(a0, b0, acc[0][0]);
            acc[0][1] = wmma4(a0, b1, acc[0][1]);
            acc[1][0] = wmma4(a1, b0, acc[1][0]);
            acc[1][1] = wmma4(a1, b1, acc[1][1]);
        }
    };

#if HAVE_ASYNC_LDS
    stage(0, 0);
    for (int kc = 0; kc < NN; kc += 16) {
        int cur = (kc >> 4) & 1;
        if (kc + 16 < NN) {
            stage(kc + 16, cur ^ 1);           // prefetch next chunk
            WAIT_ASYNC(TRANS_A ? 3 : 6);       // wait only for current chunk
        } else {
            WAIT_ASYNC(0);
        }
        __syncthreads();                        // LDS visibility across waves
        compute(cur);
        __syncthreads();                        // before next-next overwrite
    }
#else
    for (int kc = 0; kc < NN; kc += 16) {
        stage(kc, 0);
        __syncthreads();
        compute(0);
        __syncthreads();
    }
#endif

#pragma unroll
    for (int i = 0; i < 2; ++i)
#pragma unroll
        for (int j = 0; j < 2; ++j) {
            int rb = m0 + wrow * 32 + i * 16;
            int cb = f0 + wcol * 32 + j * 16;
#pragma unroll
            for (int r = 0; r < 8; ++r) {
                int row = rb + r + 8 * half;
                int col = cb + lrow;
                float v = acc[i][j][r];
                if (ADD_RES) v += Rg[(size_t)row * FF + col];
                Cg[(size_t)row * FF + col] = v;
            }
        }
}

// ---------------- host-side orchestration ----------------
extern "C" void kernel_launch(void* const* d_in, const int* in_sizes, int n_in,
                              void* d_out, int out_size, void* d_ws, size_t ws_size,
                              hipStream_t stream) {
    (void)in_sizes; (void)n_in; (void)out_size; (void)ws_size;
    const float* x      = (const float*)d_in[0];   // [B,N,F]
    const int*   eidx   = (const int*)  d_in[1];   // [2,E]
    const float* ew     = (const float*)d_in[2];   // [B,E]
    const float* gamma  = (const float*)d_in[3];   // [F]
    const float* beta   = (const float*)d_in[4];   // [F]
    const float* W1     = (const float*)d_in[5];   // [N,H]
    const float* b1     = (const float*)d_in[6];   // [H]
    const float* W2     = (const float*)d_in[7];   // [H,N]
    const float* b2     = (const float*)d_in[8];   // [N]
    float*       out    = (float*)d_out;           // [B,N,F]

    // workspace layout (floats)
    float* ws = (float*)d_ws;
    const size_t MATS = (size_t)BB * NN * NN;      // 16M
    float* bufA = ws;                              // adjacency -> eigvec accum
    float* bufB = bufA + MATS;                     // L -> (sorted) eigenvectors U
    float* eig  = bufB + MATS;                     // [B,N]
    float* dis  = eig  + (size_t)BB * NN;          // [B,N]
    float* xn   = dis  + (size_t)BB * NN;          // [B,N,F]
    float* x1   = xn   + (size_t)BB * NN * FF;     // [B,N,F]
    float* hbuf = x1   + (size_t)BB * NN * FF;     // [B,H] (B rows = 16)
    float* evb  = hbuf + (size_t)BB * HH;          // [B,N]

    // 1) adjacency: zero + scatter
    zero_kernel<<<(int)((MATS + 255) / 256), 256, 0, stream>>>(bufA, (int)MATS);
    scatter_kernel<<<(BB * EE + 255) / 256, 256, 0, stream>>>(eidx, ew, bufA);

    // 2) D^{-1/2} and Laplacian (bufA -> bufB)
    degree_kernel<<<BB * NN, 256, 0, stream>>>(bufA, dis);
    build_L_kernel<<<dim3((NN * NN) / 256, BB), 256, 0, stream>>>(bufA, dis, bufB);

    // 3) batched Jacobi eigendecomposition: work on bufB, accumulate V in bufA,
    //    sorted eigenvectors written back to bufB (=U), eigenvalues to eig.
    jacobi_evd_kernel<<<BB, 1024, 0, stream>>>(bufB, bufA, bufB, eig);

    // 4) LayerNorm (residual xn kept; relu fused into gemm1 fragment reads)
    layernorm_kernel<<<BB * NN, 256, 0, stream>>>(x, gamma, beta, xn);

    // 5) eigenvalue MLP: h = relu(eig @ W1 + b1); ev = h @ W2 + b2
    mlp_gemm_kernel<true ><<<HH / 16, 32, 0, stream>>>(eig,  W1, b1, hbuf, NN, HH);
    mlp_gemm_kernel<false><<<NN / 16, 32, 0, stream>>>(hbuf, W2, b2, evb,  HH, NN);

    // 6) spectral filter: x1 = U^T relu(xn)
    gemm_wmma_kernel<true, true, false, false>
        <<<dim3(NN / 64, FF / 128, BB), 256, 0, stream>>>(bufB, xn, nullptr, nullptr, x1);

    // 7) out = U (x1 * ev) + xn
    gemm_wmma_kernel<false, false, true, true>
        <<<dim3(NN / 64, FF / 128, BB), 256, 0, stream>>>(bufB, x1, evb, xn, out);
}